// EdgeModule_68066641707591
// MI455X (gfx1250) — compile-verified
//
#include <hip/hip_runtime.h>
#include <hip/hip_bf16.h>

typedef __attribute__((ext_vector_type(2))) float v2f;
typedef __attribute__((ext_vector_type(8))) float v8f;

#define IN_FEATS 64

// ---------------------------------------------------------------------------
// Pass 1: node-side GEMM  s_u[n] = h[n].W_u ; s_v[n] = h[n].W_v
// One wave (32 lanes) computes a 16-node tile via V_WMMA_F32_16X16X4_F32,
// accumulating over K = 64 in 16 steps of K=4.
//
// A (16x4 f32) layout (ISA 7.12.2): lanes 0-15: M=lane, v0=K0,v1=K1;
//                                   lanes 16-31: M=lane-16, v0=K2,v1=K3.
// B (4x16 f32) built symmetrically: lane n holds column n of rows K0..K1
//   (lanes 16-31: rows K2..K3). Only N=0 (W_u) and N=1 (W_v) are nonzero;
//   built branchlessly (clamped address + select) so EXEC stays all-1s
//   through the WMMA chain with no saveexec churn.
// C/D (16x16 f32): VGPR r: lanes 0-15 -> (M=r, N=lane); lanes 16-31 ->
//   (M=8+r, N=lane-16). We harvest N=0,1 from lanes 0,1,16,17.
// ---------------------------------------------------------------------------
__global__ void __launch_bounds__(32)
node_gemv_wmma(const float* __restrict__ h,
               const float* __restrict__ W,   // W[0..63]=W_u, W[64..127]=W_v
               float* __restrict__ su,
               float* __restrict__ sv,
               int n_nodes) {
  const int lane    = threadIdx.x;          // 0..31, single wave32
  const int m       = lane & 15;            // row within tile / column N for B
  const int khalf   = lane >> 4;            // 0: K pair {0,1}; 1: K pair {2,3}
  const int rowbase = blockIdx.x * 16;
  const int row     = min(rowbase + m, n_nodes - 1);
  const float* hrow = h + (size_t)row * IN_FEATS;

  // Branchless preload of all 16 B fragments. Every lane reads a valid
  // address (W is 512B, L1-resident); lanes with N>=2 are zeroed by `sel`.
  const float  sel   = (m < 2) ? 1.0f : 0.0f;
  const float* wbase = W + (m & 1) * IN_FEATS;  // N=0 -> W_u, N=1 -> W_v
  v2f Bfrag[16];
#pragma unroll
  for (int kk = 0; kk < 16; ++kk) {
    const v2f wv = *(const v2f*)(wbase + kk * 4 + khalf * 2);
    Bfrag[kk].x = wv.x * sel;
    Bfrag[kk].y = wv.y * sel;
  }

  v8f acc = (v8f)(0.0f);
#pragma unroll
  for (int kk = 0; kk < 16; ++kk) {
    const int k0 = kk * 4 + khalf * 2;
    // A fragment: two consecutive K elements of this lane's row (8B load).
    const v2f A = *(const v2f*)(hrow + k0);
    // D = A(16x4) * B(4x16) + C   -> v_wmma_f32_16x16x4_f32
    acc = __builtin_amdgcn_wmma_f32_16x16x4_f32(
        /*neg_a=*/false, A, /*neg_b=*/false, Bfrag[kk],
        /*c_mod=*/(short)0, acc, /*reuse_a=*/false, /*reuse_b=*/false);
  }

  // Harvest columns N=0 (s_u) and N=1 (s_v).
  if (m < 2) {
    float* outp   = (m == 0) ? su : sv;
    const int off = rowbase + khalf * 8;      // lanes 0,1 -> M=0..7; 16,17 -> M=8..15
#pragma unroll
    for (int r = 0; r < 8; ++r) {
      const int n = off + r;
      if (n < n_nodes) outp[n] = acc[r];
    }
  }
}

// ---------------------------------------------------------------------------
// Pass 2: per-edge gather of the two precomputed scores + bias + sigmoid.
// 4 edges per thread: int4 index loads, 4-byte gathers into the 400 KB
// (L2-resident) score tables, float4 coalesced store.
// ---------------------------------------------------------------------------
__device__ __forceinline__ float sigmoidf_fast(float x) {
  return 1.0f / (1.0f + __expf(-x));
}

__global__ void __launch_bounds__(256)
edge_logit_sigmoid(const int* __restrict__ src,
                   const int* __restrict__ dst,
                   const float* __restrict__ su,
                   const float* __restrict__ sv,
                   const float* __restrict__ bias_p,
                   float* __restrict__ out,
                   int n_edges) {
  const int t = blockIdx.x * blockDim.x + threadIdx.x;
  const int e = t * 4;
  if (e >= n_edges) return;

  const float bias = bias_p[0];

  if (e + 3 < n_edges) {
    const int4 s4 = *(const int4*)(src + e);
    const int4 d4 = *(const int4*)(dst + e);
    float4 o;
    o.x = sigmoidf_fast(su[s4.x] + sv[d4.x] + bias);
    o.y = sigmoidf_fast(su[s4.y] + sv[d4.y] + bias);
    o.z = sigmoidf_fast(su[s4.z] + sv[d4.z] + bias);
    o.w = sigmoidf_fast(su[s4.w] + sv[d4.w] + bias);
    *(float4*)(out + e) = o;
  } else {
    for (int i = e; i < n_edges; ++i)
      out[i] = sigmoidf_fast(su[src[i]] + sv[dst[i]] + bias);
  }
}

// ---------------------------------------------------------------------------
// Launch: pass 1 (node GEMM into workspace), then pass 2 (edge pass).
// Stream ordering provides the inter-pass dependency.
// ---------------------------------------------------------------------------
extern "C" void kernel_launch(void* const* d_in, const int* in_sizes, int n_in,
                              void* d_out, int out_size, void* d_ws, size_t ws_size,
                              hipStream_t stream) {
  const float* h    = (const float*)d_in[0];
  const int*   src  = (const int*)d_in[1];   // harness: integer -> const int*
  const int*   dst  = (const int*)d_in[2];
  const float* W    = (const float*)d_in[3];
  const float* bias = (const float*)d_in[4];
  float*       out  = (float*)d_out;

  const int n_nodes = in_sizes[0] / IN_FEATS;
  const int n_edges = in_sizes[1];

  float* su = (float*)d_ws;                  // n_nodes floats
  float* sv = su + n_nodes;                  // n_nodes floats (400 KB total)

  const int tiles = (n_nodes + 15) / 16;
  node_gemv_wmma<<<tiles, 32, 0, stream>>>(h, W, su, sv, n_nodes);

  const int nthreads = (n_edges + 3) / 4;
  const int blocks   = (nthreads + 255) / 256;
  edge_logit_sigmoid<<<blocks, 256, 0, stream>>>(src, dst, su, sv, bias, out, n_edges);
}